// TracerModel_38972533244652
// MI455X (gfx1250) — compile-verified
//
#include <hip/hip_runtime.h>

// Cart-pole single Euler step, B = 4,194,304 independent elements.
// Memory-bound streaming kernel: ~52 B traffic per element vs ~35 VALU ops.
// Strategy (MI455X / gfx1250):
//   - wave32, 256-thread blocks, 4 elements per thread
//   - all traffic as 128-bit vector ops (global_load_b128 / global_store_b128)
//   - nontemporal (TH_NT) hints: 218 MB single-pass stream > 192 MB L2
//   - params read via uniform scalar loads (s_load), broadcast to the wave
//   - native v_sin_f32 / v_cos_f32 / v_rcp_f32 to keep VALU << HBM time

typedef __attribute__((ext_vector_type(4))) float v4f;

struct ParamPtrs {
    const float *a1, *a2, *a4, *a5, *a6, *a7, *a8, *a10, *a11, *a12, *a13, *a14;
    const float *dt, *g, *sx, *sv, *st, *so;
};

struct ParamVals {
    float a1, a2, a4, a5, a6, a7, a8, a10, a11, a12, a13, a14;
    float dt, g, sx, sv, st, so;
};

__device__ __forceinline__ v4f step_one(float x, float v, float th, float om, float u,
                                        float n1, float n2, float n3, float n4,
                                        const ParamVals& p) {
    // native hardware transcendentals: v_sin_f32 / v_cos_f32
    float s  = __sinf(th);
    float c  = __cosf(th);
    float gs = p.g * s;
    float c2 = c * c;
    float om2s = om * om * s;

    // v_dot = (a1*u - a14*v + a2*om^2*s - a4*g*s*c) / (a6 - a7*c^2)
    float inv_dv = __builtin_amdgcn_rcpf(fmaf(-p.a7, c2, p.a6));   // v_rcp_f32
    float num_v  = fmaf(p.a1, u,
                   fmaf(-p.a14, v,
                   fmaf(p.a2, om2s, -(p.a4 * gs * c))));
    float v_dot  = num_v * inv_dv;

    // om_dot = (a10*g*s - a8*c*u - a5*om^2*s*c - a11*om - a13*om*c) / (a12 - a7*c^2)
    float inv_do = __builtin_amdgcn_rcpf(fmaf(-p.a7, c2, p.a12));
    float num_o  = fmaf(p.a10, gs,
                   fmaf(-p.a8, c * u,
                   fmaf(-p.a5, om2s * c,
                   fmaf(-p.a11, om, -(p.a13 * om * c)))));
    float om_dot = num_o * inv_do;

    v4f r;
    r.x = fmaf(p.sx, n1, fmaf(p.dt, v,      x));   // x_n
    r.y = fmaf(p.sv, n2, fmaf(p.dt, v_dot,  v));   // v_n
    r.z = fmaf(p.st, n3, fmaf(p.dt, om,     th));  // th_n
    r.w = fmaf(p.so, n4, fmaf(p.dt, om_dot, om));  // om_n
    return r;
}

__device__ __forceinline__ v4f nt_load4(const float* __restrict__ p) {
    return __builtin_nontemporal_load(reinterpret_cast<const v4f*>(p));
}

__global__ __launch_bounds__(256)
void cartpole_step_kernel(const float* __restrict__ Xp,  const float* __restrict__ Xv,
                          const float* __restrict__ Xt,  const float* __restrict__ Xo,
                          const float* __restrict__ U,
                          const float* __restrict__ N1,  const float* __restrict__ N2,
                          const float* __restrict__ N3,  const float* __restrict__ N4,
                          float* __restrict__ out, int n, ParamPtrs pp) {
    // Uniform (wave-invariant) param loads -> scalar loads, broadcast.
    ParamVals p;
    p.a1 = *pp.a1;  p.a2 = *pp.a2;  p.a4 = *pp.a4;  p.a5 = *pp.a5;
    p.a6 = *pp.a6;  p.a7 = *pp.a7;  p.a8 = *pp.a8;  p.a10 = *pp.a10;
    p.a11 = *pp.a11; p.a12 = *pp.a12; p.a13 = *pp.a13; p.a14 = *pp.a14;
    p.dt = *pp.dt;  p.g = *pp.g;
    p.sx = *pp.sx;  p.sv = *pp.sv;  p.st = *pp.st;  p.so = *pp.so;

    const int base = (blockIdx.x * blockDim.x + threadIdx.x) * 4;

    if (base + 4 <= n) {
        // 9x global_load_b128 (nontemporal) per thread, fully coalesced.
        v4f x  = nt_load4(Xp + base);
        v4f v  = nt_load4(Xv + base);
        v4f th = nt_load4(Xt + base);
        v4f om = nt_load4(Xo + base);
        v4f u  = nt_load4(U  + base);
        v4f n1 = nt_load4(N1 + base);
        v4f n2 = nt_load4(N2 + base);
        v4f n3 = nt_load4(N3 + base);
        v4f n4 = nt_load4(N4 + base);

#pragma unroll
        for (int k = 0; k < 4; ++k) {
            v4f r = step_one(x[k], v[k], th[k], om[k], u[k],
                             n1[k], n2[k], n3[k], n4[k], p);
            // Output layout [B,4,1]: element e's 4 results are contiguous
            // -> one global_store_b128 per element, coalesced across lanes.
            __builtin_nontemporal_store(
                r, reinterpret_cast<v4f*>(out + 4 * (base + k)));
        }
    } else {
        // Tail (not hit for B = 4Mi, kept for generality).
        for (int k = 0; k < 4; ++k) {
            const int i = base + k;
            if (i < n) {
                v4f r = step_one(Xp[i], Xv[i], Xt[i], Xo[i], U[i],
                                 N1[i], N2[i], N3[i], N4[i], p);
                __builtin_nontemporal_store(
                    r, reinterpret_cast<v4f*>(out + 4 * i));
            }
        }
    }
}

extern "C" void kernel_launch(void* const* d_in, const int* in_sizes, int n_in,
                              void* d_out, int out_size, void* d_ws, size_t ws_size,
                              hipStream_t stream) {
    (void)out_size; (void)d_ws; (void)ws_size;

    const float* Xp = (const float*)d_in[0];
    const float* Xv = (const float*)d_in[1];
    const float* Xt = (const float*)d_in[2];
    const float* Xo = (const float*)d_in[3];
    const float* U  = (const float*)d_in[4];
    const float* N1 = (const float*)d_in[5];
    const float* N2 = (const float*)d_in[6];
    const float* N3 = (const float*)d_in[7];
    const float* N4 = (const float*)d_in[8];

    ParamPtrs pp;
    if (n_in >= 27) {
        // params flattened as 18 single-element arrays in _PARAM_VALS
        // insertion order: a1,a2,a4,a5,a6,a7,a8,a10,a11,a12,a13,a14,dt,g,
        //                  sigma_x,sigma_v,sigma_theta,sigma_omega
        pp.a1  = (const float*)d_in[9];   pp.a2  = (const float*)d_in[10];
        pp.a4  = (const float*)d_in[11];  pp.a5  = (const float*)d_in[12];
        pp.a6  = (const float*)d_in[13];  pp.a7  = (const float*)d_in[14];
        pp.a8  = (const float*)d_in[15];  pp.a10 = (const float*)d_in[16];
        pp.a11 = (const float*)d_in[17];  pp.a12 = (const float*)d_in[18];
        pp.a13 = (const float*)d_in[19];  pp.a14 = (const float*)d_in[20];
        pp.dt  = (const float*)d_in[21];  pp.g   = (const float*)d_in[22];
        pp.sx  = (const float*)d_in[23];  pp.sv  = (const float*)d_in[24];
        pp.st  = (const float*)d_in[25];  pp.so  = (const float*)d_in[26];
    } else {
        // fallback: params packed as one 18-element array (same order)
        const float* pb = (const float*)d_in[9];
        pp.a1 = pb + 0;  pp.a2 = pb + 1;  pp.a4 = pb + 2;  pp.a5 = pb + 3;
        pp.a6 = pb + 4;  pp.a7 = pb + 5;  pp.a8 = pb + 6;  pp.a10 = pb + 7;
        pp.a11 = pb + 8; pp.a12 = pb + 9; pp.a13 = pb + 10; pp.a14 = pb + 11;
        pp.dt = pb + 12; pp.g = pb + 13;
        pp.sx = pb + 14; pp.sv = pb + 15; pp.st = pb + 16; pp.so = pb + 17;
    }

    const int n = in_sizes[0];                 // B elements
    const int threads = 256;                   // 8 wave32 per block
    const int ept = 4;                         // elements per thread
    const int blocks = (n + threads * ept - 1) / (threads * ept);  // 4096 for B=4Mi

    cartpole_step_kernel<<<blocks, threads, 0, stream>>>(
        Xp, Xv, Xt, Xo, U, N1, N2, N3, N4, (float*)d_out, n, pp);
}